// Model_81209241633220
// MI455X (gfx1250) — compile-verified
//
#include <hip/hip_runtime.h>
#include <hip/hip_bf16.h>
#include <float.h>

// Row-wise top-8 of a 4096 x 8192 fp32 matrix.
// One wave32 per row: each lane scans 256 elements (64 x float4 coalesced
// loads, 512B per load instruction across the wave), keeps a register-resident
// sorted top-8, then the wave merges 32 lists with 8 rounds of a 5-step
// __shfl_xor argmax reduction (wave32 cross-lane ops -> ds_swizzle/permlane
// on gfx1250). Memory-bound: 128 MiB read (~5.5us at 23.3 TB/s HBM; fits in
// the 192MB L2 so replays run from L2). Regular temporal loads on purpose.

#define TOPK_ROWS 4096
#define TOPK_COLS 8192
#define TOPK_K    8

__global__ __launch_bounds__(256) void topk8_rowwise_kernel(
    const float* __restrict__ x,
    float* __restrict__ out_vals,
    int*   __restrict__ out_idx) {

  const int lane = threadIdx.x & 31;     // wave32 lane
  const int wave = threadIdx.x >> 5;     // 8 waves per block
  const int row  = blockIdx.x * 8 + wave;
  if (row >= TOPK_ROWS) return;

  const float* __restrict__ rp = x + (size_t)row * TOPK_COLS;

  // Per-lane sorted top-8 (descending), kept entirely in registers.
  float v[TOPK_K];
  int   id[TOPK_K];
#pragma unroll
  for (int j = 0; j < TOPK_K; ++j) {
    v[j]  = -FLT_MAX;
    id[j] = 0x7fffffff;
  }

  // 8192 cols / (32 lanes * 4 per float4) = 64 iterations per lane.
  constexpr int ITERS = TOPK_COLS / (32 * 4);
#pragma unroll 4
  for (int t = 0; t < ITERS; ++t) {
    const int col = (t * 32 + lane) * 4;

    // Speculative prefetch a few chunks ahead (global_prefetch_b8 on gfx1250).
    if (t + 8 < ITERS) {
      __builtin_prefetch(rp + (size_t)((t + 8) * 32 + lane) * 4, 0, 3);
    }

    const float4 d = *reinterpret_cast<const float4*>(rp + col);
    float e[4] = {d.x, d.y, d.z, d.w};

#pragma unroll
    for (int q = 0; q < 4; ++q) {
      float cv = e[q];
      // Fast reject: most elements are <= current 8th best.
      if (cv > v[TOPK_K - 1]) {
        int ci = col + q;
        // Bubble the new element down the sorted list (strict > keeps
        // earlier-index elements ahead on ties -> matches lax.top_k).
#pragma unroll
        for (int j = 0; j < TOPK_K; ++j) {
          if (cv > v[j]) {
            float tv = v[j]; int ti = id[j];
            v[j] = cv;       id[j] = ci;
            cv = tv;         ci = ti;
          }
        }
      }
    }
  }

  // Merge 32 per-lane sorted lists into the row's global top-8.
  // Each round: wave-wide argmax of the lane heads (value desc, index asc on
  // tie), owning lane pops its head, lane 0 records the winner.
#pragma unroll
  for (int k = 0; k < TOPK_K; ++k) {
    float bv = v[0];
    int   bi = id[0];
#pragma unroll
    for (int off = 16; off > 0; off >>= 1) {
      float ov = __shfl_xor(bv, off, 32);
      int   oi = __shfl_xor(bi, off, 32);
      if (ov > bv || (ov == bv && oi < bi)) {
        bv = ov;
        bi = oi;
      }
    }
    // Indices are unique across lanes, so bi identifies the owning lane.
    if (bi == id[0]) {
#pragma unroll
      for (int j = 0; j < TOPK_K - 1; ++j) {
        v[j]  = v[j + 1];
        id[j] = id[j + 1];
      }
      v[TOPK_K - 1]  = -FLT_MAX;
      id[TOPK_K - 1] = 0x7fffffff;
    }
    if (lane == 0) {
      out_vals[(size_t)row * TOPK_K + k] = bv;
      out_idx [(size_t)row * TOPK_K + k] = bi;
    }
  }
}

extern "C" void kernel_launch(void* const* d_in, const int* in_sizes, int n_in,
                              void* d_out, int out_size, void* d_ws, size_t ws_size,
                              hipStream_t stream) {
  (void)in_sizes; (void)n_in; (void)d_ws; (void)ws_size; (void)out_size;

  const float* x = (const float*)d_in[0];

  // d_out layout (tuple concatenated flat in return order):
  //   [0, 4096*8)          : values, fp32
  //   [4096*8, 2*4096*8)   : indices, int32 (raw bits)
  float* out_vals = (float*)d_out;
  int*   out_idx  = (int*)d_out + (size_t)TOPK_ROWS * TOPK_K;

  const dim3 block(256);                 // 8 wave32s per block -> 8 rows
  const dim3 grid(TOPK_ROWS / 8);        // 512 blocks, 4096 waves total

  topk8_rowwise_kernel<<<grid, block, 0, stream>>>(x, out_vals, out_idx);
}